// Columbina_Model_42760694399650
// MI455X (gfx1250) — compile-verified
//
#include <hip/hip_runtime.h>
#include <hip/hip_bf16.h>
#include <cstdint>

typedef float v2f __attribute__((ext_vector_type(2)));
typedef float v8f __attribute__((ext_vector_type(8)));

// ---------------------------------------------------------------------------
// helpers
// ---------------------------------------------------------------------------
__device__ __forceinline__ unsigned f2ord(float f) {
    unsigned u = __float_as_uint(f);
    return (u & 0x80000000u) ? ~u : (u | 0x80000000u);
}
__device__ __forceinline__ float ord2f(unsigned u) {
    return __uint_as_float((u & 0x80000000u) ? (u ^ 0x80000000u) : ~u);
}

__global__ void fill_f32_kernel(float* __restrict__ p, float v, long n) {
    long i = (long)blockIdx.x * blockDim.x + threadIdx.x;
    if (i < n) p[i] = v;
}
__global__ void fill_u32_kernel(unsigned* __restrict__ p, unsigned v, long n) {
    long i = (long)blockIdx.x * blockDim.x + threadIdx.x;
    if (i < n) p[i] = v;
}

// ---------------------------------------------------------------------------
// Encoder: Y = relu(X @ W + b), accumulate per-channel sum / sumsq for BN.
// One wave per 16-row tile; N = 32 output channels = two 16x16 WMMA N-tiles.
// fp32 WMMA 16x16x4.
// ---------------------------------------------------------------------------
__global__ __launch_bounds__(32) void encode_wmma_kernel(
    const float* __restrict__ X, const float* __restrict__ W,
    const float* __restrict__ bias, float* __restrict__ Y,
    float* __restrict__ bnsum, float* __restrict__ bnsq,
    int N, int Kin)
{
    const int tile = blockIdx.x;
    const int lane = threadIdx.x;
    const int col  = lane & 15;           // N within 16-wide tile
    const int koff = (lane >> 4) * 2;     // K pair selected by lane half
    int rowA = tile * 16 + col;           // A row = lane&15
    if (rowA >= N) rowA = N - 1;          // clamp (keeps EXEC uniform)

    v8f c0 = {}; v8f c1 = {};
    for (int kk = 0; kk < Kin; kk += 4) {
        const int k0 = kk + koff;
        v2f a;  a[0]  = X[(size_t)rowA * Kin + k0];
                a[1]  = X[(size_t)rowA * Kin + k0 + 1];
        v2f b0; b0[0] = W[(size_t)k0 * 32 + col];
                b0[1] = W[(size_t)(k0 + 1) * 32 + col];
        v2f b1; b1[0] = W[(size_t)k0 * 32 + 16 + col];
                b1[1] = W[(size_t)(k0 + 1) * 32 + 16 + col];
        c0 = __builtin_amdgcn_wmma_f32_16x16x4_f32(false, a, false, b0, (short)0, c0, false, false);
        c1 = __builtin_amdgcn_wmma_f32_16x16x4_f32(false, a, false, b1, (short)0, c1, false, false);
    }

    const int mbase = (lane >> 4) * 8;    // C layout: lanes16-31 hold rows M+8
    float s0 = 0.f, q0 = 0.f, s1 = 0.f, q1 = 0.f;
    const float bi0 = bias[col], bi1 = bias[col + 16];
#pragma unroll
    for (int r = 0; r < 8; ++r) {
        const int row = tile * 16 + mbase + r;
        float v0 = c0[r] + bi0; v0 = v0 > 0.f ? v0 : 0.f;
        float v1 = c1[r] + bi1; v1 = v1 > 0.f ? v1 : 0.f;
        if (row < N) {
            Y[(size_t)row * 32 + col]      = v0;
            Y[(size_t)row * 32 + col + 16] = v1;
            s0 += v0; q0 += v0 * v0;
            s1 += v1; q1 += v1 * v1;
        }
    }
    atomicAdd(&bnsum[col],      s0); atomicAdd(&bnsq[col],      q0);
    atomicAdd(&bnsum[col + 16], s1); atomicAdd(&bnsq[col + 16], q1);
}

__global__ void bn_finalize_kernel(const float* __restrict__ bnsum,
                                   const float* __restrict__ bnsq,
                                   const float* __restrict__ gamma,
                                   const float* __restrict__ beta,
                                   float* __restrict__ scale,
                                   float* __restrict__ shift, float invN)
{
    int c = threadIdx.x;
    if (c < 32) {
        float mean = bnsum[c] * invN;
        float var  = bnsq[c] * invN - mean * mean;
        float sc   = gamma[c] * rsqrtf(var + 1e-5f);
        scale[c] = sc;
        shift[c] = beta[c] - mean * sc;
    }
}

__global__ void bn_apply_kernel(float* __restrict__ Y,
                                const float* __restrict__ scale,
                                const float* __restrict__ shift, long n)
{
    long i = (long)blockIdx.x * blockDim.x + threadIdx.x;
    if (i < n) {
        int c = (int)(i & 31);
        Y[i] = Y[i] * scale[c] + shift[c];
    }
}

// ---------------------------------------------------------------------------
// Source projection: H = ENC @ Wbig, ENC [N,32], Wbig [32,128] -> H [N,128].
// One wave per 16-row tile, 8 N-tiles -> 64 WMMAs per wave.
// ---------------------------------------------------------------------------
__global__ __launch_bounds__(32) void hproj_wmma_kernel(
    const float* __restrict__ ENC, const float* __restrict__ Wb,
    float* __restrict__ H, int N)
{
    const int tile = blockIdx.x;
    const int lane = threadIdx.x;
    const int col  = lane & 15;
    const int koff = (lane >> 4) * 2;
    int rowA = tile * 16 + col;
    if (rowA >= N) rowA = N - 1;

    v8f c[8];
#pragma unroll
    for (int t = 0; t < 8; ++t) c[t] = (v8f){};

#pragma unroll
    for (int kk = 0; kk < 32; kk += 4) {
        const int k0 = kk + koff;
        v2f a; a[0] = ENC[(size_t)rowA * 32 + k0];
               a[1] = ENC[(size_t)rowA * 32 + k0 + 1];
#pragma unroll
        for (int t = 0; t < 8; ++t) {
            v2f b; b[0] = Wb[(size_t)k0 * 128 + t * 16 + col];
                   b[1] = Wb[(size_t)(k0 + 1) * 128 + t * 16 + col];
            c[t] = __builtin_amdgcn_wmma_f32_16x16x4_f32(false, a, false, b, (short)0, c[t], false, false);
        }
    }

    const int mbase = (lane >> 4) * 8;
#pragma unroll
    for (int r = 0; r < 8; ++r) {
        const int row = tile * 16 + mbase + r;
        if (row < N) {
#pragma unroll
            for (int t = 0; t < 8; ++t)
                H[(size_t)row * 128 + t * 16 + col] = c[t][r];
        }
    }
}

// ---------------------------------------------------------------------------
// Fold attention vector into W:  P[k,h] = sum_c W[k, h*32+c] * att[h,c]
// so that a[i,h] = enc[i,:] . P[:,h]  (avoids materializing x@W for scores).
// ---------------------------------------------------------------------------
__global__ void proj_att_kernel(const float* __restrict__ W,
                                const float* __restrict__ att,
                                float* __restrict__ P)
{
    int t = threadIdx.x;       // 128 threads: k = t>>2 (0..31), h = t&3
    if (t < 128) {
        int k = t >> 2, hh = t & 3;
        float s = 0.f;
#pragma unroll
        for (int cc = 0; cc < 32; ++cc)
            s += W[(size_t)k * 128 + hh * 32 + cc] * att[hh * 32 + cc];
        P[k * 4 + hh] = s;
    }
}

__global__ void compute_a_kernel(const float* __restrict__ enc,
                                 const float* __restrict__ P,
                                 float* __restrict__ A, int N)
{
    int t = blockIdx.x * blockDim.x + threadIdx.x;
    if (t < N * 4) {
        int i = t >> 2, hh = t & 3;
        float s = 0.f;
#pragma unroll
        for (int k = 0; k < 32; ++k)
            s += enc[(size_t)i * 32 + k] * P[k * 4 + hh];
        A[t] = s;
    }
}

// ---------------------------------------------------------------------------
// Edge passes (softmax over incoming edges, then weighted scatter).
// Thread layout: tid -> (edge e, head hh). Self-loops appended at e >= E.
// ---------------------------------------------------------------------------
__device__ __forceinline__ void edge_decode(int e, int E, const int* __restrict__ src,
                                            const int* __restrict__ dst, int& s, int& d)
{
    if (e < E) { s = src[e]; d = dst[e]; }
    else       { s = e - E;  d = e - E; }
}

__global__ void edge_max_kernel(const int* __restrict__ src, const int* __restrict__ dst,
                                int E, int Nloop,
                                const float* __restrict__ a_s, const float* __restrict__ a_d,
                                unsigned* __restrict__ m_enc)
{
    int tid = blockIdx.x * blockDim.x + threadIdx.x;
    int tot = (E + Nloop) * 4;
    if (tid >= tot) return;
    int e = tid >> 2, hh = tid & 3, s, d;
    edge_decode(e, E, src, dst, s, d);
    float al = a_s[s * 4 + hh] + a_d[d * 4 + hh];
    al = al > 0.f ? al : 0.2f * al;                 // leaky_relu(0.2)
    atomicMax(&m_enc[d * 4 + hh], f2ord(al));
}

__global__ void edge_sum_kernel(const int* __restrict__ src, const int* __restrict__ dst,
                                int E, int Nloop,
                                const float* __restrict__ a_s, const float* __restrict__ a_d,
                                const unsigned* __restrict__ m_enc,
                                float* __restrict__ denom)
{
    int tid = blockIdx.x * blockDim.x + threadIdx.x;
    int tot = (E + Nloop) * 4;
    if (tid >= tot) return;
    int e = tid >> 2, hh = tid & 3, s, d;
    edge_decode(e, E, src, dst, s, d);
    float al = a_s[s * 4 + hh] + a_d[d * 4 + hh];
    al = al > 0.f ? al : 0.2f * al;
    float ev = __expf(al - ord2f(m_enc[d * 4 + hh]));
    atomicAdd(&denom[d * 4 + hh], ev);
}

__global__ void edge_scatter_kernel(const int* __restrict__ src, const int* __restrict__ dst,
                                    int E, int Nloop,
                                    const float* __restrict__ a_s, const float* __restrict__ a_d,
                                    const unsigned* __restrict__ m_enc,
                                    const float* __restrict__ denom,
                                    const float* __restrict__ H,
                                    float* __restrict__ acc)
{
    int tid = blockIdx.x * blockDim.x + threadIdx.x;
    int tot = (E + Nloop) * 4;
    if (tid >= tot) return;
    int e = tid >> 2, hh = tid & 3, s, d;
    edge_decode(e, E, src, dst, s, d);
    float al = a_s[s * 4 + hh] + a_d[d * 4 + hh];
    al = al > 0.f ? al : 0.2f * al;
    float ev = __expf(al - ord2f(m_enc[d * 4 + hh]));
    float a  = ev / (denom[d * 4 + hh] + 1e-16f);
    const float* hs = H   + (size_t)s * 128 + hh * 32;
    float*       ac = acc + (size_t)d * 128 + hh * 32;
#pragma unroll
    for (int cc = 0; cc < 32; ++cc)
        atomicAdd(&ac[cc], hs[cc] * a);
}

// out[i,c] = gin[i,c] + mean_heads(acc[i,:,c]) + bias[c]   (gin may alias out)
__global__ void finalize_kernel(const float* __restrict__ gin,
                                const float* __restrict__ acc,
                                const float* __restrict__ bias,
                                float* __restrict__ out, long n)
{
    long t = (long)blockIdx.x * blockDim.x + threadIdx.x;
    if (t < n) {
        long i = t >> 5; int cc = (int)(t & 31);
        const float* a = acc + i * 128;
        float m = 0.25f * (a[cc] + a[32 + cc] + a[64 + cc] + a[96 + cc]);
        out[t] = gin[t] + m + bias[cc];
    }
}

// ---------------------------------------------------------------------------
// Host launcher
// ---------------------------------------------------------------------------
static inline unsigned nblk(long n, int b) { return (unsigned)((n + b - 1) / b); }

extern "C" void kernel_launch(void* const* d_in, const int* in_sizes, int n_in,
                              void* d_out, int out_size, void* d_ws, size_t ws_size,
                              hipStream_t stream)
{
    const float* x_gene  = (const float*)d_in[0];
    const float* x_drug  = (const float*)d_in[1];
    const int*   edge_gg = (const int*)d_in[2];
    const int*   edge_dg = (const int*)d_in[3];
    const float* Wg      = (const float*)d_in[4];
    const float* bg      = (const float*)d_in[5];
    const float* gg_gamma= (const float*)d_in[6];
    const float* gg_beta = (const float*)d_in[7];
    const float* Wd      = (const float*)d_in[8];
    const float* bd      = (const float*)d_in[9];
    const float* dg_gamma= (const float*)d_in[10];
    const float* dg_beta = (const float*)d_in[11];
    const float* W1      = (const float*)d_in[12];
    const float* att_s1  = (const float*)d_in[13];
    const float* att_d1  = (const float*)d_in[14];
    const float* bias1   = (const float*)d_in[15];
    const float* W2      = (const float*)d_in[16];
    const float* att_s2  = (const float*)d_in[17];
    const float* att_d2  = (const float*)d_in[18];
    const float* bias2   = (const float*)d_in[19];

    const int NG = in_sizes[0] / 128;
    const int ND = in_sizes[1] / 96;
    const int E1 = in_sizes[2] / 2;
    const int E2 = in_sizes[3] / 2;
    const int NM = NG > ND ? NG : ND;
    const int NLOOP2 = NG < ND ? NG : ND;

    size_t off = 0;
    auto alloc = [&](size_t nbytes) -> void* {
        void* p = (char*)d_ws + off;
        off += (nbytes + 255) & ~(size_t)255;
        return p;
    };
    float*    g_enc = (float*)   alloc((size_t)NG * 32 * 4);
    float*    d_enc = (float*)   alloc((size_t)ND * 32 * 4);
    float*    hbuf  = (float*)   alloc((size_t)NM * 128 * 4);
    float*    acc   = (float*)   alloc((size_t)NG * 128 * 4);
    float*    a_s   = (float*)   alloc((size_t)NM * 4 * 4);
    float*    a_d   = (float*)   alloc((size_t)NG * 4 * 4);
    unsigned* m_enc = (unsigned*)alloc((size_t)NG * 4 * 4);
    float*    denom = (float*)   alloc((size_t)NG * 4 * 4);
    float*    bnsum = (float*)   alloc(32 * 4);
    float*    bnsq  = (float*)   alloc(32 * 4);
    float*    scale = (float*)   alloc(32 * 4);
    float*    shift = (float*)   alloc(32 * 4);
    float*    P     = (float*)   alloc(128 * 4);
    (void)ws_size; (void)n_in; (void)out_size;

    float* gout = (float*)d_out;

    const int  TB = 256;
    const long ngc = (long)NG * 32;
    const long ndc = (long)ND * 32;

    // ---- gene encoder: Linear+ReLU, BN ----
    fill_f32_kernel<<<1, 64, 0, stream>>>(bnsum, 0.f, 32);
    fill_f32_kernel<<<1, 64, 0, stream>>>(bnsq,  0.f, 32);
    encode_wmma_kernel<<<nblk(NG, 16), 32, 0, stream>>>(x_gene, Wg, bg, g_enc, bnsum, bnsq, NG, 128);
    bn_finalize_kernel<<<1, 32, 0, stream>>>(bnsum, bnsq, gg_gamma, gg_beta, scale, shift, 1.0f / (float)NG);
    bn_apply_kernel<<<nblk(ngc, TB), TB, 0, stream>>>(g_enc, scale, shift, ngc);

    // ---- drug encoder ----
    fill_f32_kernel<<<1, 64, 0, stream>>>(bnsum, 0.f, 32);
    fill_f32_kernel<<<1, 64, 0, stream>>>(bnsq,  0.f, 32);
    encode_wmma_kernel<<<nblk(ND, 16), 32, 0, stream>>>(x_drug, Wd, bd, d_enc, bnsum, bnsq, ND, 96);
    bn_finalize_kernel<<<1, 32, 0, stream>>>(bnsum, bnsq, dg_gamma, dg_beta, scale, shift, 1.0f / (float)ND);
    bn_apply_kernel<<<nblk(ndc, TB), TB, 0, stream>>>(d_enc, scale, shift, ndc);

    // ===================== GAT 1: Gene -> Gene =====================
    proj_att_kernel<<<1, 128, 0, stream>>>(W1, att_s1, P);
    compute_a_kernel<<<nblk((long)NG * 4, TB), TB, 0, stream>>>(g_enc, P, a_s, NG);
    proj_att_kernel<<<1, 128, 0, stream>>>(W1, att_d1, P);
    compute_a_kernel<<<nblk((long)NG * 4, TB), TB, 0, stream>>>(g_enc, P, a_d, NG);
    hproj_wmma_kernel<<<nblk(NG, 16), 32, 0, stream>>>(g_enc, W1, hbuf, NG);

    fill_u32_kernel<<<nblk((long)NG * 4, TB), TB, 0, stream>>>(m_enc, 0u, (long)NG * 4);
    fill_f32_kernel<<<nblk((long)NG * 4, TB), TB, 0, stream>>>(denom, 0.f, (long)NG * 4);
    fill_f32_kernel<<<nblk((long)NG * 128, TB), TB, 0, stream>>>(acc, 0.f, (long)NG * 128);

    {
        const int* s1 = edge_gg;
        const int* d1 = edge_gg + E1;
        const long tot = (long)(E1 + NG) * 4;
        edge_max_kernel    <<<nblk(tot, TB), TB, 0, stream>>>(s1, d1, E1, NG, a_s, a_d, m_enc);
        edge_sum_kernel    <<<nblk(tot, TB), TB, 0, stream>>>(s1, d1, E1, NG, a_s, a_d, m_enc, denom);
        edge_scatter_kernel<<<nblk(tot, TB), TB, 0, stream>>>(s1, d1, E1, NG, a_s, a_d, m_enc, denom, hbuf, acc);
    }
    finalize_kernel<<<nblk(ngc, TB), TB, 0, stream>>>(g_enc, acc, bias1, gout, ngc);

    // ===================== GAT 2: Drug -> Gene =====================
    proj_att_kernel<<<1, 128, 0, stream>>>(W2, att_s2, P);
    compute_a_kernel<<<nblk((long)ND * 4, TB), TB, 0, stream>>>(d_enc, P, a_s, ND);
    proj_att_kernel<<<1, 128, 0, stream>>>(W2, att_d2, P);
    compute_a_kernel<<<nblk((long)NG * 4, TB), TB, 0, stream>>>(gout, P, a_d, NG);   // x_dst = updated g
    hproj_wmma_kernel<<<nblk(ND, 16), 32, 0, stream>>>(d_enc, W2, hbuf, ND);

    fill_u32_kernel<<<nblk((long)NG * 4, TB), TB, 0, stream>>>(m_enc, 0u, (long)NG * 4);
    fill_f32_kernel<<<nblk((long)NG * 4, TB), TB, 0, stream>>>(denom, 0.f, (long)NG * 4);
    fill_f32_kernel<<<nblk((long)NG * 128, TB), TB, 0, stream>>>(acc, 0.f, (long)NG * 128);

    {
        const int* s2 = edge_dg;
        const int* d2 = edge_dg + E2;
        const long tot = (long)(E2 + NLOOP2) * 4;
        edge_max_kernel    <<<nblk(tot, TB), TB, 0, stream>>>(s2, d2, E2, NLOOP2, a_s, a_d, m_enc);
        edge_sum_kernel    <<<nblk(tot, TB), TB, 0, stream>>>(s2, d2, E2, NLOOP2, a_s, a_d, m_enc, denom);
        edge_scatter_kernel<<<nblk(tot, TB), TB, 0, stream>>>(s2, d2, E2, NLOOP2, a_s, a_d, m_enc, denom, hbuf, acc);
    }
    finalize_kernel<<<nblk(ngc, TB), TB, 0, stream>>>(gout, acc, bias2, gout, ngc);  // in-place residual
}